// mk_84859963834781
// MI455X (gfx1250) — compile-verified
//
#include <hip/hip_runtime.h>
#include <hip/hip_bf16.h>

// MI455X / gfx1250 (CDNA5, wave32) implementation.
//
// Per batch element b:
//   nn_dt = (n^T diag(w) n) / dt          (n shared 13x10)
//   nxnx  = nn_dt + k*(nx1^T W nx1 + nx2^T W nx2)
//   r1    = b_bc + nn_dt*c_n - nxnx*c_i
//   diagA = diag(nxnx)
//
// One wave32 per element. Each 10x10 Gram product is a padded 16x16 WMMA
// with K contracted in 4-wide chunks using V_WMMA_F32_16X16X4_F32:
//   A[M=i, K=g] = src[g,i]            (lane = M, K-pair chosen by lane>=16)
//   B[K=g, N=j] = src[g,j] * w[g]*c   (lane = N, same K-pair rule)
// 12 WMMAs per element (4 for nn_dt, 4+4 for the two nx Grams).
//
// Padding strategy (branch-free, EXEC stays all-1s for WMMA):
//  - K rows g>=13: zeroed ONLY via the preloaded B-scales.
//  - Columns j>=10: junk in C; consumers are masked instead (cn/ci zeroed,
//    row results selected only on lanes with j==m<10).
//  - Rows m>=10 of C: never stored.
//
// Epilogue: row sums reduced with DPP ROW_XMASK adds (no LDS, no waits);
// per-row results selected into lanes 0..9 (one ds_swizzle SWAPX16 to move
// rows 8,9 across the half-wave) and stored with two coalesced b32 stores.

typedef __attribute__((ext_vector_type(2))) float v2f;
typedef __attribute__((ext_vector_type(8))) float v8f;

// x + lane(x ^ MASK) within each DPP row of 16 lanes (gfx10+ ROW_XMASK)
#define XOR_ADD(x, MASK)                                                      \
    ((x) + __int_as_float(__builtin_amdgcn_update_dpp(                        \
               0, __float_as_int(x), 0x160 | (MASK), 0xF, 0xF, true)))

__device__ __forceinline__ void accum_phase(v8f& acc,
                                            const float* __restrict__ mat, // 13x10 row-major (g,i)
                                            const float (&s)[8],           // per-lane K-scales (0 for g>=13)
                                            int jm, int half)
{
    #pragma unroll
    for (int ch = 0; ch < 4; ++ch) {
        const int g0 = ch * 4 + half * 2;
        const int g1 = g0 + 1;
        // chunks 0..2 have g <= 11 (statically in range); only chunk 3 clamps
        const int g0c = (ch == 3) ? (g0 > 12 ? 12 : g0) : g0;
        const int g1c = (ch == 3) ? (g1 > 12 ? 12 : g1) : g1;

        const float a0 = mat[g0c * 10 + jm];
        const float a1 = mat[g1c * 10 + jm];

        v2f A  = {a0, a1};
        v2f Bf = {a0 * s[2 * ch], a1 * s[2 * ch + 1]};
        // 8 args: (neg_a, A, neg_b, B, c_mod, C, reuse_a, reuse_b)
        acc = __builtin_amdgcn_wmma_f32_16x16x4_f32(false, A, false, Bf,
                                                    (short)0, acc, false, false);
    }
}

extern "C" __global__ void __launch_bounds__(256)
fem_assemble_wmma(const float* __restrict__ c_i,
                  const float* __restrict__ c_n,
                  const float* __restrict__ b_bc,
                  const float* __restrict__ kptr,
                  const float* __restrict__ dtptr,
                  const float* __restrict__ nmat,
                  const float* __restrict__ nx,
                  const float* __restrict__ detwei,
                  float* __restrict__ out_diag,
                  float* __restrict__ out_r1,
                  int Btot)
{
    const int lane = threadIdx.x & 31;
    const int b    = blockIdx.x * 8 + (threadIdx.x >> 5); // 8 waves per 256-thread block
    if (b >= Btot) return;                                // wave-uniform exit

    const int j    = lane & 15;
    const int jm   = (j > 9) ? 9 : j;   // clamped column index
    const int half = lane >> 4;

    const float kk  = kptr[0];
    const float rdt = 1.0f / dtptr[0];

    // Preload this lane's 8 quadrature weights once. The g>=13 padding mask
    // lives HERE (only chunk-3 entries can be out of range), pre-scaled for
    // both phase types.
    const size_t b13 = (size_t)b * 13;
    float s_dt[8], s_k[8];
    #pragma unroll
    for (int ch = 0; ch < 4; ++ch) {
        const int g0  = ch * 4 + half * 2;
        const int g1  = g0 + 1;
        const int g0c = (ch == 3) ? (g0 > 12 ? 12 : g0) : g0;
        const int g1c = (ch == 3) ? (g1 > 12 ? 12 : g1) : g1;
        float w0 = detwei[b13 + g0c];
        float w1 = detwei[b13 + g1c];
        if (ch == 3) {            // fold the K-padding mask into the scales
            w0 *= (g0 < 13) ? 1.0f : 0.0f;
            w1 *= (g1 < 13) ? 1.0f : 0.0f;
        }
        s_dt[2 * ch]     = w0 * rdt;
        s_dt[2 * ch + 1] = w1 * rdt;
        s_k[2 * ch]      = w0 * kk;
        s_k[2 * ch + 1]  = w1 * kk;
    }

    v8f acc = {0.f, 0.f, 0.f, 0.f, 0.f, 0.f, 0.f, 0.f};

    // Phase 1: nn_dt (shared basis matrix, per-element weights)
    accum_phase(acc, nmat, s_dt, jm, half);
    const v8f c_nn = acc;  // snapshot nn/dt

    // Phases 2-3: accumulate k-scaled nx Grams on top -> nxnx
    const float* __restrict__ nx1 = nx + (size_t)b * 260;
    const float* __restrict__ nx2 = nx1 + 130;
    accum_phase(acc, nx1, s_k, jm, half);
    accum_phase(acc, nx2, s_k, jm, half);

    // Epilogue. Lane holds column j of rows m = v (+8 for upper half-wave).
    // Columns j>=10 of C hold junk -> zero cn/ci so those lanes contribute 0.
    const size_t b10 = (size_t)b * 10;
    const float jmf = (j < 10) ? 1.0f : 0.0f;
    const float cn  = c_n[b10 + jm] * jmf;
    const float ci  = c_i[b10 + jm] * jmf;

    float dsel = 0.0f, tsel = 0.0f;   // per-lane row results (diag, r-sum)
    #pragma unroll
    for (int v = 0; v < 8; ++v) {
        const int   m  = (lane < 16) ? v : v + 8;
        const float cv = acc[v];

        // row-dot: nn_dt[m,:]*c_n - nxnx[m,:]*c_i, reduced across the 16-lane
        // DPP row; afterwards every lane in the row holds the full sum.
        float t = c_nn[v] * cn - cv * ci;
        t = XOR_ADD(t, 1);
        t = XOR_ADD(t, 2);
        t = XOR_ADD(t, 4);
        t = XOR_ADD(t, 8);

        // The diagonal lane of row m (j == m) keeps both results in registers.
        const bool pick = (j == m);
        dsel = pick ? cv : dsel;
        tsel = pick ? t  : tsel;
    }

    // Rows 0..7 now sit in lanes 0..7; rows 8..15 sit in lanes 24..31.
    // Swap the 16-lane halves (ds_swizzle SWAPX16: xor_mask=0x10) so lanes
    // 8,9 receive rows 8,9, then store coalesced from lanes 0..9.
    const float dsw = __int_as_float(
        __builtin_amdgcn_ds_swizzle(__float_as_int(dsel), 0x401F));
    const float tsw = __int_as_float(
        __builtin_amdgcn_ds_swizzle(__float_as_int(tsel), 0x401F));
    dsel = (lane >= 8) ? dsw : dsel;
    tsel = (lane >= 8) ? tsw : tsel;

    if (lane < 10) {
        out_diag[b10 + lane] = dsel;
        out_r1[b10 + lane]   = b_bc[b10 + lane] + tsel;
    }
}

extern "C" void kernel_launch(void* const* d_in, const int* in_sizes, int n_in,
                              void* d_out, int out_size, void* d_ws, size_t ws_size,
                              hipStream_t stream)
{
    const float* c_i    = (const float*)d_in[0]; // (B,1,10)
    const float* c_n    = (const float*)d_in[1]; // (B,1,10)
    const float* b_bc   = (const float*)d_in[2]; // (B,10)
    const float* k      = (const float*)d_in[3]; // (1,)
    const float* dt     = (const float*)d_in[4]; // (1,)
    const float* nmat   = (const float*)d_in[5]; // (13,10)
    const float* nx     = (const float*)d_in[6]; // (B,2,13,10)
    const float* detwei = (const float*)d_in[7]; // (B,13)

    const int B = in_sizes[2] / 10; // from b_bc

    float* out_diag = (float*)d_out;              // (B,10)
    float* out_r1   = out_diag + (size_t)B * 10;  // (B,1,10) flattened after diagA

    dim3 block(256);                // 8 wave32 per block, one element per wave
    dim3 grid((B + 7) / 8);
    hipLaunchKernelGGL(fem_assemble_wmma, grid, block, 0, stream,
                       c_i, c_n, b_bc, k, dt, nmat, nx, detwei,
                       out_diag, out_r1, B);
}